// NUFTBase_6554120094041
// MI455X (gfx1250) — compile-verified
//
#include <hip/hip_runtime.h>
#include <math.h>

typedef __attribute__((ext_vector_type(2))) float v2f;
typedef __attribute__((ext_vector_type(8))) float v8f;

#define KPTS   32768
#define NCH    32          // B*C = 8*4
#define GXY    256
#define HW     128
#define ALPHA  14.04f      // 2.34 * J, J=6
#define TWOPI  6.28318530717958647692f

// ---------------------------------------------------------------- Bessel I0
__device__ __forceinline__ float bessel_i0f(float x) {
    float ax = fabsf(x);
    if (ax < 3.75f) {
        float t = x / 3.75f; t *= t;
        return 1.0f + t*(3.5156229f + t*(3.0899424f + t*(1.2067492f +
                     t*(0.2659732f + t*(0.0360768f + t*0.0045813f)))));
    } else {
        float t = 3.75f / ax;
        return (expf(ax) * rsqrtf(ax)) *
               (0.39894228f + t*(0.01328592f + t*(0.00225319f +
                t*(-0.00157565f + t*(0.00916281f + t*(-0.02057706f +
                t*(0.02635537f + t*(-0.01647633f + t*0.00392377f))))))));
    }
}

// apodization 1D factor: 1 / FT(kb) at centered coord i (i in [0,128))
__device__ __forceinline__ float apod1d(int i, float i0a) {
    float om = ((float)i - 63.5f) * (1.0f / 256.0f);
    float t  = 3.14159265358979f * 6.0f * om;
    float s2 = ALPHA * ALPHA - t * t;      // always > 0 in this range
    float s  = sqrtf(s2);
    float kb = 6.0f * (sinhf(s) / s) / i0a;
    return 1.0f / kb;
}

// ------------------------------------------------- table builder (DFT mats)
__global__ void build_tables(float* __restrict__ Ar,  float* __restrict__ Ai,
                             float* __restrict__ Art, float* __restrict__ Ait,
                             float* __restrict__ Bsr, float* __restrict__ Bsi,
                             float* __restrict__ Bstr,float* __restrict__ Bsti,
                             float* __restrict__ apod) {
    int tid = blockIdx.x * blockDim.x + threadIdx.x;   // 32768 threads
    // A[h,u] = exp(+2*pi*i*h*u/256), 128x256 ; At = transpose (256x128)
    {
        int h = tid >> 8, u = tid & 255;
        float th = TWOPI * (float)((h * u) & 255) * (1.0f / 256.0f);
        float c = cosf(th), s = sinf(th);
        Ar [h * 256 + u] = c;  Ai [h * 256 + u] = s;
        Art[u * 128 + h] = c;  Ait[u * 128 + h] = s;
    }
    if (tid < 128 * 128) {
        int p = tid >> 7, h = tid & 127;
        // Bs[p,h] = (-1)^h * exp(-2*pi*i*p*h/128)  (fftshift baked in)
        float th  = TWOPI * (float)((p * h) & 127) * (1.0f / 128.0f);
        float sgn = (h & 1) ? -1.0f : 1.0f;
        float c = sgn * cosf(th), s = -sgn * sinf(th);
        Bsr [p * 128 + h] = c;  Bsi [p * 128 + h] = s;
        Bstr[h * 128 + p] = c;  Bsti[h * 128 + p] = s;  // plain transpose
        // apodization (outer product of 1D factors)
        float i0a = bessel_i0f(ALPHA);
        apod[tid] = apod1d(p, i0a) * apod1d(h, i0a);
    }
}

// ------------------------------------------------- per-point KB weights
__global__ void compute_weights(const float* __restrict__ coords,
                                float* __restrict__ wx, float* __restrict__ wy,
                                int* __restrict__ kxi, int* __restrict__ kyi) {
    int k = blockIdx.x * blockDim.x + threadIdx.x;
    if (k >= KPTS) return;
    float i0a = bessel_i0f(ALPHA);
    float scale = (float)GXY / TWOPI;
    float gx = coords[2 * k + 0] * scale;
    float gy = coords[2 * k + 1] * scale;
    float bx = floorf(gx - 3.0f);
    float by = floorf(gy - 3.0f);
#pragma unroll
    for (int j = 0; j < 6; ++j) {
        float kfx = bx + (float)(j + 1);
        float ux  = gx - kfx;
        float ax  = 1.0f - (ux * (1.0f / 3.0f)) * (ux * (1.0f / 3.0f));
        float wxx = (ax > 0.0f)
                  ? bessel_i0f(ALPHA * sqrtf(fmaxf(ax, 1e-12f))) / i0a : 0.0f;
        wx[k * 6 + j] = wxx;
        int ki = (int)kfx;
        kxi[k * 6 + j] = ((ki % GXY) + GXY) % GXY;

        float kfy = by + (float)(j + 1);
        float uy  = gy - kfy;
        float ay  = 1.0f - (uy * (1.0f / 3.0f)) * (uy * (1.0f / 3.0f));
        float wyy = (ay > 0.0f)
                  ? bessel_i0f(ALPHA * sqrtf(fmaxf(ay, 1e-12f))) / i0a : 0.0f;
        wy[k * 6 + j] = wyy;
        int kj = (int)kfy;
        kyi[k * 6 + j] = ((kj % GXY) + GXY) % GXY;
    }
}

// ------------------------------------------------- gridding, LDS-privatized
// One 1024-thread (32-wave) workgroup per channel: full 256KB grid plane
// lives in LDS (CDNA5: 320KB/WGP). ds_add_f32 atomics replace 37.7M global
// atomics; plane is written back with plain coalesced stores (no global
// atomics, no separate zeroing pass).
__global__ void grid_scatter_lds(const float* __restrict__ values,
                                 const float* __restrict__ wx,
                                 const float* __restrict__ wy,
                                 const int* __restrict__ kxi,
                                 const int* __restrict__ kyi,
                                 float* __restrict__ grid) {
    extern __shared__ float lds[];            // GXY*GXY floats = 256 KB
    int ch = blockIdx.x;                      // channel plane 0..31
    int t  = threadIdx.x;                     // 0..1023
    for (int i = t; i < GXY * GXY; i += 1024) lds[i] = 0.0f;
    __syncthreads();

    const float* vch = values + ch * KPTS;
    for (int k = t; k < KPTS; k += 1024) {
        float v = vch[k];
        float wxr[6], wyr[6];
        int   kxr[6], kyr[6];
#pragma unroll
        for (int j = 0; j < 6; ++j) {
            wxr[j] = wx[k * 6 + j];
            wyr[j] = wy[k * 6 + j];
            kxr[j] = kxi[k * 6 + j] << 8;     // row offset
            kyr[j] = kyi[k * 6 + j];
        }
#pragma unroll
        for (int i2 = 0; i2 < 6; ++i2) {
            float vw = v * wxr[i2];
#pragma unroll
            for (int j = 0; j < 6; ++j)
                atomicAdd(&lds[kxr[i2] + kyr[j]], vw * wyr[j]);
        }
    }
    __syncthreads();

    float* g = grid + ch * (GXY * GXY);
    for (int i = t; i < GXY * GXY; i += 1024) g[i] = lds[i];
}

// ------------------------------------------------- complex GEMM via WMMA f32
// C = (Ar + i*Ai) @ (Br + i*Bi). Bi may be null (real B).
// Epilogues: apod != null -> elementwise scale; absOut != null -> |C| with
// batch/channel fftshift roll.
__global__ void cgemm_wmma(const float* __restrict__ Ar, const float* __restrict__ Ai, int sA,
                           const float* __restrict__ Br, const float* __restrict__ Bi, int sB,
                           float* __restrict__ Cr, float* __restrict__ Ci, int sC,
                           const float* __restrict__ apod, float* __restrict__ absOut,
                           int M, int N, int K) {
    int img  = blockIdx.y;
    int wave = blockIdx.x * (blockDim.x >> 5) + (threadIdx.x >> 5);
    int tn_count = N >> 4;
    if (wave >= (M >> 4) * tn_count) return;          // wave-uniform
    int tm = (wave / tn_count) << 4;
    int tn = (wave % tn_count) << 4;
    int lane = threadIdx.x & 31;
    int l    = lane & 15;
    int hi   = lane >> 4;

    const float* arp = Ar + (long)sA * img;
    const float* aip = Ai + (long)sA * img;
    const float* brp = Br + (long)sB * img;
    const float* bip = Bi ? (Bi + (long)sB * img) : nullptr;

    v8f cr = {};
    v8f ci = {};
    int arow = tm + l;                                 // A rows: M = lane&15
    for (int k0 = 0; k0 < K; k0 += 4) {
        int ka = k0 + 2 * hi;                          // half-lanes split K
        v2f a_r, a_i, b_r;
        a_r.x = arp[arow * K + ka];      a_r.y = arp[arow * K + ka + 1];
        a_i.x = aip[arow * K + ka];      a_i.y = aip[arow * K + ka + 1];
        b_r.x = brp[ka * N + tn + l];    b_r.y = brp[(ka + 1) * N + tn + l];
        cr = __builtin_amdgcn_wmma_f32_16x16x4_f32(false, a_r, false, b_r,
                                                   (short)0, cr, false, false);
        ci = __builtin_amdgcn_wmma_f32_16x16x4_f32(false, a_i, false, b_r,
                                                   (short)0, ci, false, false);
        if (bip) {
            v2f b_i, na_i;
            b_i.x = bip[ka * N + tn + l]; b_i.y = bip[(ka + 1) * N + tn + l];
            na_i.x = -a_i.x;              na_i.y = -a_i.y;   // f32 WMMA has no A-neg
            cr = __builtin_amdgcn_wmma_f32_16x16x4_f32(false, na_i, false, b_i,
                                                       (short)0, cr, false, false);
            ci = __builtin_amdgcn_wmma_f32_16x16x4_f32(false, a_r, false, b_i,
                                                       (short)0, ci, false, false);
        }
    }

    if (absOut) {
        // fftshift over batch(8)/channel(4) dims: out[b,c] = src[(b+4)%8,(c+2)%4]
        int b = img >> 2, c = img & 3;
        int dst = (((b + 4) & 7) << 2) | ((c + 2) & 3);
        float* outp = absOut + (long)dst * (HW * HW);
#pragma unroll
        for (int r = 0; r < 8; ++r) {
            int m = tm + r + hi * 8, n = tn + l;
            float vr = cr[r], vi = ci[r];
            outp[m * N + n] = sqrtf(vr * vr + vi * vi);
        }
    } else {
        float* crp = Cr + (long)sC * img;
        float* cip = Ci + (long)sC * img;
#pragma unroll
        for (int r = 0; r < 8; ++r) {
            int m = tm + r + hi * 8, n = tn + l;
            float vr = cr[r], vi = ci[r];
            if (apod) { float ap = apod[m * HW + n]; vr *= ap; vi *= ap; }
            crp[m * N + n] = vr;
            cip[m * N + n] = vi;
        }
    }
}

// ------------------------------------------------------------------ launch
extern "C" void kernel_launch(void* const* d_in, const int* in_sizes, int n_in,
                              void* d_out, int out_size, void* d_ws, size_t ws_size,
                              hipStream_t stream) {
    const float* values = (const float*)d_in[0];   // [8,4,32768]
    const float* coords = (const float*)d_in[1];   // [32768,2]
    float* out = (float*)d_out;                    // [8,4,128,128]
    float* ws  = (float*)d_ws;

    // workspace carve (floats); ~29 MB total
    float* Ar   = ws;              float* Ai   = Ar   + 32768;
    float* Art  = Ai   + 32768;    float* Ait  = Art  + 32768;
    float* Bsr  = Ait  + 32768;    float* Bsi  = Bsr  + 16384;
    float* Bstr = Bsi  + 16384;    float* Bsti = Bstr + 16384;
    float* apod = Bsti + 16384;
    float* wx   = apod + 16384;    float* wy   = wx   + KPTS * 6;
    int*   kxi  = (int*)(wy + KPTS * 6);
    int*   kyi  = kxi + KPTS * 6;
    float* grid = (float*)(kyi + KPTS * 6);        // 32 * 65536
    float* Pr   = grid + NCH * GXY * GXY;          // 32 * 128*256
    float* Pi   = Pr   + NCH * HW * GXY;
    float* imgR = Pi   + NCH * HW * GXY;           // 32 * 128*128
    float* imgI = imgR + NCH * HW * HW;
    float* Rr   = imgI + NCH * HW * HW;
    float* Ri   = Rr   + NCH * HW * HW;

    build_tables   <<<128, 256, 0, stream>>>(Ar, Ai, Art, Ait,
                                             Bsr, Bsi, Bstr, Bsti, apod);
    compute_weights<<<128, 256, 0, stream>>>(coords, wx, wy, kxi, kyi);

    // one block per channel; full plane accumulated in 256KB of LDS
    grid_scatter_lds<<<NCH, 1024, GXY * GXY * sizeof(float), stream>>>(
        values, wx, wy, kxi, kyi, grid);

    // Stage 1: P = A @ grid          (128x256 = [128x256]@[256x256], real B)
    cgemm_wmma<<<dim3(16, NCH), 256, 0, stream>>>(
        Ar, Ai, 0,  grid, nullptr, GXY * GXY,  Pr, Pi, HW * GXY,
        nullptr, nullptr, 128, 256, 256);
    // Stage 2: img = apod * (P @ A^T)     (128x128 = [128x256]@[256x128])
    cgemm_wmma<<<dim3(8, NCH), 256, 0, stream>>>(
        Pr, Pi, HW * GXY,  Art, Ait, 0,  imgR, imgI, HW * HW,
        apod, nullptr, 128, 128, 256);
    // Stage 3: R = Bs @ img
    cgemm_wmma<<<dim3(8, NCH), 256, 0, stream>>>(
        Bsr, Bsi, 0,  imgR, imgI, HW * HW,  Rr, Ri, HW * HW,
        nullptr, nullptr, 128, 128, 128);
    // Stage 4: out = |R @ Bs^T|  (with batch/channel roll)
    cgemm_wmma<<<dim3(8, NCH), 256, 0, stream>>>(
        Rr, Ri, HW * HW,  Bstr, Bsti, 0,  nullptr, nullptr, 0,
        nullptr, out, 128, 128, 128);
}